// LargestConnectedComponent_15006615732922
// MI455X (gfx1250) — compile-verified
//
#include <hip/hip_runtime.h>
#include <stdint.h>

#define BATCH 16
#define NCLS 4
#define HH 768
#define WW 768
#define HW (HH * WW)                 // 589824 (divisible by 1024 and 256)
#define NPLANES (BATCH * (NCLS - 1)) // 48

// ---------------------------------------------------------------------------
// Kernel 1: channel argmax. Stage 4 channel slabs (4KB each) into LDS with the
// gfx1250 async global->LDS DMA path (ASYNCcnt), then compute per-pixel argmax.
// First-max tie-breaking (strict >) matches jnp.argmax.
// ---------------------------------------------------------------------------
__global__ __launch_bounds__(256) void k_argmax(const float* __restrict__ pred,
                                                unsigned char* __restrict__ lbl8) {
  __shared__ float tile[NCLS][1024];
  const int t = threadIdx.x;
  const int block_pix = blockIdx.x * 1024;   // 1024 divides HW -> never straddles images
  const int b = block_pix / HW;
  const int base = block_pix - b * HW;
  const float* img = pred + (size_t)b * NCLS * HW + base;

#pragma unroll
  for (int c = 0; c < NCLS; ++c) {
    // Low 32 bits of a flat pointer into __shared__ are the LDS byte address.
    unsigned lds_addr = (unsigned)(unsigned long long)(&tile[c][t * 4]);
    unsigned long long gaddr = (unsigned long long)(img + (size_t)c * HW + t * 4);
    asm volatile("global_load_async_to_lds_b128 %0, %1, off"
                 :: "v"(lds_addr), "v"(gaddr) : "memory");
  }
  asm volatile("s_wait_asynccnt 0x0" ::: "memory");
  __syncthreads();

#pragma unroll
  for (int i = 0; i < 4; ++i) {
    const int p = t * 4 + i;
    float best = tile[0][p];
    int bi = 0;
#pragma unroll
    for (int c = 1; c < NCLS; ++c) {
      const float v = tile[c][p];
      if (v > best) { best = v; bi = c; }
    }
    lbl8[(size_t)b * HW + base + p] = (unsigned char)bi;
  }
}

// ---------------------------------------------------------------------------
// Concurrent union-find (ECL-CC style). Links always point to smaller indices,
// so the converged root is the minimum pixel index of the component — exactly
// the reference's min-label flood-fill fixed point.
// ---------------------------------------------------------------------------
__device__ __forceinline__ int ufind(const int* __restrict__ L, int i) {
  int r = L[i];
  int p = L[r];
  while (r != p) { r = p; p = L[r]; }
  return r;
}

__device__ __forceinline__ void uunite(int* L, int a, int b) {
  int ra = ufind(L, a);
  int rb = ufind(L, b);
  while (ra != rb) {
    if (ra < rb) { int tmp = ra; ra = rb; rb = tmp; }  // ra > rb
    const int old = atomicCAS(&L[ra], ra, rb);
    if (old == ra) break;   // hooked larger root under smaller
    ra = old;               // ra's parent changed; retry against it
  }
}

__global__ __launch_bounds__(256) void k_init(int* __restrict__ L,
                                              int* __restrict__ counts,
                                              unsigned long long* __restrict__ best) {
  const int idx = blockIdx.x * blockDim.x + threadIdx.x;
  if (idx >= NPLANES * HW) return;
  const int plane = idx / HW;
  const int p = idx - plane * HW;
  L[idx] = p;        // every pixel its own root (bg pixels stay singletons)
  counts[idx] = 0;   // size histogram lives in d_out scratch; re-zeroed each call
  if (idx < NPLANES) best[idx] = 0ull;
}

__global__ __launch_bounds__(256) void k_merge(const unsigned char* __restrict__ lbl8,
                                               int* __restrict__ L) {
  const int idx = blockIdx.x * blockDim.x + threadIdx.x;
  if (idx >= NPLANES * HW) return;
  const int plane = idx / HW;
  const int p = idx - plane * HW;
  const int b = plane / (NCLS - 1);
  const int cls = plane - b * (NCLS - 1) + 1;                // classes 1..3
  const unsigned char* m = lbl8 + (size_t)b * HW;
  if ((int)m[p] != cls) return;
  int* Lp = L + (size_t)plane * HW;
  const int x = p % WW;
  if (x + 1 < WW && (int)m[p + 1] == cls)  uunite(Lp, p, p + 1);   // right edge
  if (p + WW < HW && (int)m[p + WW] == cls) uunite(Lp, p, p + WW); // down edge
}

// Fused flatten + size-count: path-compress every pixel to its (now stable)
// root and, for foreground pixels, bump the component-size histogram keyed by
// that same root. Saves a full 113 MB re-read of L versus separate kernels.
__global__ __launch_bounds__(256) void k_flatcount(const unsigned char* __restrict__ lbl8,
                                                   int* __restrict__ L,
                                                   int* __restrict__ counts) {
  const int idx = blockIdx.x * blockDim.x + threadIdx.x;
  if (idx >= NPLANES * HW) return;
  const int plane = idx / HW;
  const int p = idx - plane * HW;
  int* Lp = L + (size_t)plane * HW;
  const int root = ufind(Lp, p);
  Lp[p] = root;  // monotone-decreasing write: safe vs concurrent finds
  const int b = plane / (NCLS - 1);
  const int cls = plane - b * (NCLS - 1) + 1;
  if ((int)lbl8[(size_t)b * HW + p] == cls)
    atomicAdd(&counts[(size_t)plane * HW + root], 1);
}

// key = (size<<32) | (HW-1-root): max size wins; ties -> smallest root,
// matching jnp.argmax(sizes) first-maximum semantics.
__global__ __launch_bounds__(256) void k_best(const int* __restrict__ counts,
                                              unsigned long long* __restrict__ best) {
  const int idx = blockIdx.x * blockDim.x + threadIdx.x;
  if (idx >= NPLANES * HW) return;
  const int plane = idx / HW;
  const int p = idx - plane * HW;
  const int c = counts[idx];
  if (c > 0) {
    const unsigned long long key =
        ((unsigned long long)(unsigned)c << 32) | (unsigned)(HW - 1 - p);
    atomicMax(&best[plane], key);
  }
}

__global__ __launch_bounds__(256) void k_final(const unsigned char* __restrict__ lbl8,
                                               const int* __restrict__ L,
                                               const unsigned long long* __restrict__ best,
                                               float* __restrict__ out) {
  const int idx = blockIdx.x * blockDim.x + threadIdx.x;
  if (idx >= BATCH * HW) return;
  const int b = idx / HW;      // uniform per block (256 | HW)
  const int p = idx - b * HW;

  // Stage the 3 per-plane best keys through LDS once per block.
  __shared__ unsigned long long bk_sh[NCLS - 1];
  if (threadIdx.x < NCLS - 1)
    bk_sh[threadIdx.x] = best[b * (NCLS - 1) + (int)threadIdx.x];
  __syncthreads();

  const int cls = (int)lbl8[idx];
  float fg[NCLS - 1];
  float s = 0.f;
#pragma unroll
  for (int c = 1; c < NCLS; ++c) {
    const int plane = b * (NCLS - 1) + (c - 1);
    const unsigned long long bk = bk_sh[c - 1];
    const unsigned long long sz = bk >> 32;
    const int root = HW - 1 - (int)(unsigned)(bk & 0xffffffffu);
    const bool on = (cls == c) && (sz > 0ull) &&
                    (L[(size_t)plane * HW + p] == root);
    fg[c - 1] = on ? 1.f : 0.f;
    s += fg[c - 1];
  }
  const float bg = 1.f - (s > 1.f ? 1.f : s);  // masks disjoint -> s in {0,1}
  float* ob = out + (size_t)b * NCLS * HW + p;
  ob[0] = bg;
  ob[(size_t)1 * HW] = fg[0];
  ob[(size_t)2 * HW] = fg[1];
  ob[(size_t)3 * HW] = fg[2];
}

// ---------------------------------------------------------------------------
extern "C" void kernel_launch(void* const* d_in, const int* in_sizes, int n_in,
                              void* d_out, int out_size, void* d_ws, size_t ws_size,
                              hipStream_t stream) {
  (void)in_sizes; (void)n_in; (void)out_size; (void)ws_size;
  const float* pred = (const float*)d_in[0];
  float* out = (float*)d_out;

  // Workspace layout (needs ~123 MB):
  //   [0,                9437184)  u8  label map (argmax result)
  //   [9437184,        122683392)  i32 union-find labels, 48 planes x HW
  //   [122683392,      122683776)  u64 per-plane best (size, root) key
  unsigned char* lbl8 = (unsigned char*)d_ws;
  int* L = (int*)((char*)d_ws + (size_t)BATCH * HW);
  unsigned long long* best =
      (unsigned long long*)((char*)d_ws + (size_t)BATCH * HW + (size_t)NPLANES * HW * 4);
  // Component-size histogram (113 MB) borrows d_out; k_final overwrites it last.
  int* counts = (int*)d_out;

  const int nPlanePix = NPLANES * HW;           // 28,311,552 (divisible by 256)
  const int blk = 256;

  k_argmax   <<<BATCH * HW / 1024, 256, 0, stream>>>(pred, lbl8);
  k_init     <<<nPlanePix / blk, blk, 0, stream>>>(L, counts, best);
  k_merge    <<<nPlanePix / blk, blk, 0, stream>>>(lbl8, L);
  k_flatcount<<<nPlanePix / blk, blk, 0, stream>>>(lbl8, L, counts);
  k_best     <<<nPlanePix / blk, blk, 0, stream>>>(counts, best);
  k_final    <<<BATCH * HW / blk, blk, 0, stream>>>(lbl8, L, best, out);
}